// VQVAE_71408126263388
// MI455X (gfx1250) — compile-verified
//
#include <hip/hip_runtime.h>
#include <cstddef>

typedef __attribute__((ext_vector_type(2))) float v2f;
typedef __attribute__((ext_vector_type(8))) float v8f;

static constexpr int BZ = 8192;   // batch
static constexpr int D0 = 784;    // input dim
static constexpr int HD = 400;    // hidden dim
static constexpr int DZ = 256;    // latent dim
static constexpr int NE = 8192;   // codebook entries

#define WMMA_F32(Af, Bf, Cf) \
  __builtin_amdgcn_wmma_f32_16x16x4_f32(false, (Af), false, (Bf), (short)0, (Cf), false, false)

// ---------------------------------------------------------------------------
// ||emb_j||^2 per codebook row
// ---------------------------------------------------------------------------
__global__ __launch_bounds__(256) void embsq_kernel(const float* __restrict__ emb,
                                                    float* __restrict__ esq) {
  int j = blockIdx.x * blockDim.x + threadIdx.x;
  if (j < NE) {
    const float* e = emb + (size_t)j * DZ;
    float s = 0.0f;
#pragma unroll 8
    for (int c = 0; c < DZ; ++c) s = __builtin_fmaf(e[c], e[c], s);
    esq[j] = s;
  }
}

// ---------------------------------------------------------------------------
// C[M,N] = act(A[M,K] @ W[N,K]^T + bias[N])   (fp32 WMMA 16x16x4)
// grid: (ceil(N/64), M/128), block 256 = 8 waves; wave tile = 32x32.
// A-frag: lane loads A[row0 + lane%16][k + 2*(lane/16) .. +1]  (16x4, f32)
// B-frag: lane loads W[col0 + lane%16][k + 2*(lane/16) .. +1]  (4x16 = W^T tile)
// ACT: 0 = none, 1 = relu, 2 = sigmoid
// ---------------------------------------------------------------------------
template <int ACT>
__global__ __launch_bounds__(256) void gemm_wmma_kernel(
    const float* __restrict__ A, const float* __restrict__ W,
    const float* __restrict__ bias, float* __restrict__ C,
    int M, int N, int K) {
  const int lane = threadIdx.x & 31;
  const int wave = threadIdx.x >> 5;
  const int lr   = lane & 15;
  const int lk2  = (lane >> 4) << 1;  // 0 or 2
  const int row0 = blockIdx.y * 128 + (wave >> 1) * 32;
  const int col0 = blockIdx.x * 64 + (wave & 1) * 32;

  const float* a0 = A + (size_t)(row0 + lr) * K + lk2;
  const float* a1 = a0 + (size_t)16 * K;
  const int n0 = col0 + lr;
  const int n1 = col0 + 16 + lr;
  const bool s0 = n0 < N, s1 = n1 < N;
  // Clamp out-of-range columns to a valid row: garbage only reaches output
  // columns that are never stored (WMMA columns are independent).
  const float* w0 = W + (size_t)(s0 ? n0 : N - 1) * K + lk2;
  const float* w1 = W + (size_t)(s1 ? n1 : N - 1) * K + lk2;

  v8f acc00 = {}, acc01 = {}, acc10 = {}, acc11 = {};
#pragma unroll 8
  for (int k = 0; k < K; k += 4) {
    v2f af0 = *(const v2f*)(a0 + k);
    v2f af1 = *(const v2f*)(a1 + k);
    v2f bf0 = *(const v2f*)(w0 + k);
    v2f bf1 = *(const v2f*)(w1 + k);
    acc00 = WMMA_F32(af0, bf0, acc00);
    acc01 = WMMA_F32(af0, bf1, acc01);
    acc10 = WMMA_F32(af1, bf0, acc10);
    acc11 = WMMA_F32(af1, bf1, acc11);
  }

  const int rbase = row0 + ((lane >> 4) << 3);  // +8 for upper half-wave
  auto store_tile = [&](const v8f& acc, int roff, int cc, bool valid) {
    if (!valid) return;
    const float bv = bias[cc];
#pragma unroll
    for (int r = 0; r < 8; ++r) {
      float v = acc[r] + bv;
      if (ACT == 1) v = fmaxf(v, 0.0f);
      if (ACT == 2) v = 1.0f / (1.0f + __expf(-v));
      C[(size_t)(rbase + roff + r) * N + cc] = v;
    }
  };
  store_tile(acc00, 0, n0, s0);
  store_tile(acc01, 0, n1, s1);
  store_tile(acc10, 16, n0, s0);
  store_tile(acc11, 16, n1, s1);
}

// ---------------------------------------------------------------------------
// Fused distance GEMM + argmin:
//   idx[i] = argmin_j ( ||emb_j||^2 - 2 * z_e_i . emb_j )
// Block: 64 rows, 8 waves x 16 codes = 128 codes per sweep, 64 sweeps.
// Each wave holds 4 M-fragments -> 4 WMMAs per 8-byte codebook lane-load
// (16 FLOP/byte from L2; codebook re-read only 128x = ~1.07 GB of L2 traffic).
// z_e tile staged in LDS (row stride +2 floats: conflict-free frag reads).
// ---------------------------------------------------------------------------
__global__ __launch_bounds__(256) void vq_argmin_kernel(
    const float* __restrict__ ze, const float* __restrict__ emb,
    const float* __restrict__ esq, int* __restrict__ idx_out) {
  __shared__ float zs[64][DZ + 2];
  __shared__ float redV[8][64];
  __shared__ int   redI[8][64];

  const int lane = threadIdx.x & 31;
  const int wave = threadIdx.x >> 5;
  const int lr   = lane & 15;
  const int half = lane >> 4;
  const int row0 = blockIdx.x * 64;

  for (int t = threadIdx.x; t < 64 * DZ; t += 256) {
    int r = t >> 8, c = t & (DZ - 1);
    zs[r][c] = ze[(size_t)(row0 + r) * DZ + c];
  }
  __syncthreads();

  float bestV[4][8];
  int   bestI[4][8];
#pragma unroll
  for (int f = 0; f < 4; ++f)
#pragma unroll
    for (int r = 0; r < 8; ++r) { bestV[f][r] = 3.4e38f; bestI[f][r] = 0; }

  const float* za[4];
#pragma unroll
  for (int f = 0; f < 4; ++f) za[f] = &zs[f * 16 + lr][half * 2];

  for (int ct = 0; ct < NE; ct += 128) {
    const int code = ct + wave * 16 + lr;  // this lane's output column
    const float* ep = emb + (size_t)code * DZ + half * 2;
    v8f acc0 = {}, acc1 = {}, acc2 = {}, acc3 = {};
#pragma unroll 8
    for (int k = 0; k < DZ; k += 4) {
      v2f bf = *(const v2f*)(ep + k);
      v2f a0 = *(const v2f*)(za[0] + k);
      v2f a1 = *(const v2f*)(za[1] + k);
      v2f a2 = *(const v2f*)(za[2] + k);
      v2f a3 = *(const v2f*)(za[3] + k);
      acc0 = WMMA_F32(a0, bf, acc0);
      acc1 = WMMA_F32(a1, bf, acc1);
      acc2 = WMMA_F32(a2, bf, acc2);
      acc3 = WMMA_F32(a3, bf, acc3);
    }
    const float e = esq[code];
#pragma unroll
    for (int r = 0; r < 8; ++r) {
      float v0 = __builtin_fmaf(-2.0f, acc0[r], e);
      if (v0 < bestV[0][r]) { bestV[0][r] = v0; bestI[0][r] = code; }
      float v1 = __builtin_fmaf(-2.0f, acc1[r], e);
      if (v1 < bestV[1][r]) { bestV[1][r] = v1; bestI[1][r] = code; }
      float v2 = __builtin_fmaf(-2.0f, acc2[r], e);
      if (v2 < bestV[2][r]) { bestV[2][r] = v2; bestI[2][r] = code; }
      float v3 = __builtin_fmaf(-2.0f, acc3[r], e);
      if (v3 < bestV[3][r]) { bestV[3][r] = v3; bestI[3][r] = code; }
    }
  }

  // reduce across the 16 lanes of each half-wave (xor masks stay in-half)
#pragma unroll
  for (int f = 0; f < 4; ++f) {
#pragma unroll
    for (int r = 0; r < 8; ++r) {
      float v = bestV[f][r];
      int   i = bestI[f][r];
      for (int m = 1; m < 16; m <<= 1) {
        float ov = __shfl_xor(v, m, 32);
        int   oi = __shfl_xor(i, m, 32);
        if (ov < v || (ov == v && oi < i)) { v = ov; i = oi; }
      }
      if (lr == 0) {
        int lrow = f * 16 + r + half * 8;  // local row 0..63
        redV[wave][lrow] = v;
        redI[wave][lrow] = i;
      }
    }
  }
  __syncthreads();

  if (threadIdx.x < 64) {
    const int lrow = threadIdx.x;
    float v = redV[0][lrow];
    int   i = redI[0][lrow];
#pragma unroll
    for (int w = 1; w < 8; ++w) {
      float ov = redV[w][lrow];
      int   oi = redI[w][lrow];
      if (ov < v || (ov == v && oi < i)) { v = ov; i = oi; }
    }
    idx_out[row0 + lrow] = i;
  }
}

// ---------------------------------------------------------------------------
// z_q = emb[idx]  (one float4 per thread)
// ---------------------------------------------------------------------------
__global__ __launch_bounds__(256) void gather_kernel(const float* __restrict__ emb,
                                                     const int* __restrict__ idx,
                                                     float* __restrict__ zq) {
  int t = blockIdx.x * blockDim.x + threadIdx.x;
  int row = t >> 6;        // DZ/4 = 64 float4 per row
  int c4  = t & 63;
  if (row < BZ) {
    const float4* src = (const float4*)(emb + (size_t)idx[row] * DZ) + c4;
    float4* dst = (float4*)(zq + (size_t)row * DZ) + c4;
    *dst = *src;
  }
}

// ---------------------------------------------------------------------------
// Deterministic two-stage loss reduction (no float atomics).
// part[2b+0] = sum of t*clamp(log p) + (1-t)*clamp(log1p(-p)) over bce terms
// part[2b+1] = sum of (z_e - z_q)^2
// ---------------------------------------------------------------------------
__global__ __launch_bounds__(256) void loss_partial_kernel(
    const float* __restrict__ xr, const float* __restrict__ x,
    const float* __restrict__ ze, const float* __restrict__ zq,
    float* __restrict__ part) {
  const size_t N1 = (size_t)BZ * D0;
  const size_t N2 = (size_t)BZ * DZ;
  const size_t stride = (size_t)gridDim.x * blockDim.x;
  float s1 = 0.0f, s2 = 0.0f;
  for (size_t i = (size_t)blockIdx.x * blockDim.x + threadIdx.x; i < N1; i += stride) {
    float p = xr[i], t = x[i];
    float lp  = fmaxf(logf(p), -100.0f);
    float l1p = fmaxf(log1pf(-p), -100.0f);
    s1 += t * lp + (1.0f - t) * l1p;
  }
  for (size_t i = (size_t)blockIdx.x * blockDim.x + threadIdx.x; i < N2; i += stride) {
    float d = ze[i] - zq[i];
    s2 = __builtin_fmaf(d, d, s2);
  }
  __shared__ float sh1[256], sh2[256];
  sh1[threadIdx.x] = s1;
  sh2[threadIdx.x] = s2;
  __syncthreads();
  for (int o = 128; o > 0; o >>= 1) {
    if ((int)threadIdx.x < o) {
      sh1[threadIdx.x] += sh1[threadIdx.x + o];
      sh2[threadIdx.x] += sh2[threadIdx.x + o];
    }
    __syncthreads();
  }
  if (threadIdx.x == 0) {
    part[2 * blockIdx.x + 0] = sh1[0];
    part[2 * blockIdx.x + 1] = sh2[0];
  }
}

__global__ __launch_bounds__(256) void loss_final_kernel(const float* __restrict__ part,
                                                         int nblocks,
                                                         float* __restrict__ out3) {
  __shared__ float sh1[256], sh2[256];
  float s1 = 0.0f, s2 = 0.0f;
  for (int i = threadIdx.x; i < nblocks; i += 256) {
    s1 += part[2 * i + 0];
    s2 += part[2 * i + 1];
  }
  sh1[threadIdx.x] = s1;
  sh2[threadIdx.x] = s2;
  __syncthreads();
  for (int o = 128; o > 0; o >>= 1) {
    if ((int)threadIdx.x < o) {
      sh1[threadIdx.x] += sh1[threadIdx.x + o];
      sh2[threadIdx.x] += sh2[threadIdx.x + o];
    }
    __syncthreads();
  }
  if (threadIdx.x == 0) {
    out3[0] = -sh1[0] / (float)((size_t)BZ * D0);  // reconst_loss (mean BCE)
    float ec = sh2[0] / 8192.0f;
    out3[1] = ec;  // embed_loss
    out3[2] = ec;  // commit_loss (same forward value)
  }
}

// ---------------------------------------------------------------------------
extern "C" void kernel_launch(void* const* d_in, const int* in_sizes, int n_in,
                              void* d_out, int out_size, void* d_ws, size_t ws_size,
                              hipStream_t stream) {
  const float* x   = (const float*)d_in[0];
  const float* W1  = (const float*)d_in[1];
  const float* b1  = (const float*)d_in[2];
  const float* W2  = (const float*)d_in[3];
  const float* b2  = (const float*)d_in[4];
  const float* W3  = (const float*)d_in[5];
  const float* b3  = (const float*)d_in[6];
  const float* W4  = (const float*)d_in[7];
  const float* b4  = (const float*)d_in[8];
  const float* emb = (const float*)d_in[9];
  float* out = (float*)d_out;

  // workspace layout (floats): ~43 MB total
  float* ws  = (float*)d_ws;
  float* h1  = ws;                          // BZ*HD
  float* ze  = h1 + (size_t)BZ * HD;        // BZ*DZ
  float* zq  = ze + (size_t)BZ * DZ;        // BZ*DZ
  float* h3  = zq + (size_t)BZ * DZ;        // BZ*HD
  float* esq = h3 + (size_t)BZ * HD;        // NE
  int*   idx = (int*)(esq + NE);            // BZ
  float* part = (float*)(idx + BZ);         // 2*1024

  const dim3 blk(256);
  constexpr int LOSS_BLOCKS = 1024;

  embsq_kernel<<<dim3((NE + 255) / 256), blk, 0, stream>>>(emb, esq);

  // h1 = relu(x @ W1^T + b1) : M=8192 N=400 K=784
  gemm_wmma_kernel<1><<<dim3((HD + 63) / 64, BZ / 128), blk, 0, stream>>>(
      x, W1, b1, h1, BZ, HD, D0);
  // z_e = h1 @ W2^T + b2 : M=8192 N=256 K=400
  gemm_wmma_kernel<0><<<dim3((DZ + 63) / 64, BZ / 128), blk, 0, stream>>>(
      h1, W2, b2, ze, BZ, DZ, HD);

  vq_argmin_kernel<<<dim3(BZ / 64), blk, 0, stream>>>(ze, emb, esq, idx);
  gather_kernel<<<dim3((BZ * (DZ / 4)) / 256), blk, 0, stream>>>(emb, idx, zq);

  // h3 = relu(z_q @ W3^T + b3) : M=8192 N=400 K=256
  gemm_wmma_kernel<1><<<dim3((HD + 63) / 64, BZ / 128), blk, 0, stream>>>(
      zq, W3, b3, h3, BZ, HD, DZ);
  // x_reconst = sigmoid(h3 @ W4^T + b4) : M=8192 N=784 K=400 -> d_out
  gemm_wmma_kernel<2><<<dim3((D0 + 63) / 64, BZ / 128), blk, 0, stream>>>(
      h3, W4, b4, out, BZ, D0, HD);

  loss_partial_kernel<<<dim3(LOSS_BLOCKS), blk, 0, stream>>>(out, x, ze, zq, part);
  loss_final_kernel<<<dim3(1), blk, 0, stream>>>(part, LOSS_BLOCKS,
                                                 out + (size_t)BZ * D0);
}